// GraphLevelGNN_73108933312642
// MI455X (gfx1250) — compile-verified
//
#include <hip/hip_runtime.h>

#define NN 100000      // nodes
#define NE 1600000     // edges
#define F  128         // feature dim (F_IN == HID)
#define NG 256         // graphs
#define NC 10          // classes
#define MT (NN / 16)   // 6250 M-tiles, exact

typedef float v2f __attribute__((ext_vector_type(2)));
typedef float v8f __attribute__((ext_vector_type(8)));

// Hardware global_atomic_add_f32 (non-returning). unsafeAtomicAdd maps
// unconditionally to the native FP atomic on AMD; no CAS-loop expansion.
__device__ __forceinline__ void atomic_add_f32(float* p, float v) {
    unsafeAtomicAdd(p, v);
}

// ---------------- degree / norm ----------------
__global__ void k_init_deg(float* deg) {
    int i = blockIdx.x * blockDim.x + threadIdx.x;
    if (i < NN) deg[i] = 1.0f;                      // self-loop contributes 1
}

__global__ void k_deg_edges(const long long* __restrict__ dst, float* deg) {
    int e = blockIdx.x * blockDim.x + threadIdx.x;
    if (e < NE) atomic_add_f32(&deg[(int)dst[e]], 1.0f);
}

__global__ void k_dinv(float* deg) {
    int i = blockIdx.x * blockDim.x + threadIdx.x;
    if (i < NN) deg[i] = rsqrtf(deg[i]);            // deg >= 1 always (self-loops)
}

// ---------------- fp32 WMMA GEMM: C[M,128] = act(A[M,128]) @ W[128,128] ----------------
// one wave per 16-row strip; 8 C-tiles (all 128 cols) per wave; K in steps of 4.
// RELU_IN fuses the previous layer's ReLU into the A-fragment load.
template <bool RELU_IN>
__global__ __launch_bounds__(256) void k_gemm_wmma(const float* __restrict__ A,
                                                   const float* __restrict__ W,
                                                   float* __restrict__ C) {
    const int wave = threadIdx.x >> 5;
    const int lane = threadIdx.x & 31;
    const int mt   = blockIdx.x * 8 + wave;
    if (mt >= MT) return;                            // wave-uniform: EXEC stays all-1s
    const int half = lane >> 4;                      // 0: lanes 0-15, 1: lanes 16-31
    const int lm   = lane & 15;
    const float* arow = A + (size_t)(mt * 16 + lm) * F;

    v8f acc[8];
#pragma unroll
    for (int t = 0; t < 8; ++t) { v8f z = {}; acc[t] = z; }

    for (int ks = 0; ks < 32; ++ks) {
        const int k0 = ks * 4 + half * 2;            // this half-wave's K pair
        v2f a;
        a.x = arow[k0];
        a.y = arow[k0 + 1];
        if (RELU_IN) {
            a.x = fmaxf(a.x, 0.0f);
            a.y = fmaxf(a.y, 0.0f);
        }
#pragma unroll
        for (int nt = 0; nt < 8; ++nt) {
            const int col = nt * 16 + lm;
            v2f b;
            b.x = W[(size_t)k0 * F + col];
            b.y = W[(size_t)(k0 + 1) * F + col];
            acc[nt] = __builtin_amdgcn_wmma_f32_16x16x4_f32(
                false, a, false, b, (short)0, acc[nt], false, false);
        }
    }
    // C/D layout: VGPR j -> row (j + 8*half), col = lm  (16x16 f32 tile)
#pragma unroll
    for (int nt = 0; nt < 8; ++nt) {
        const int col = nt * 16 + lm;
#pragma unroll
        for (int j = 0; j < 8; ++j) {
            C[(size_t)(mt * 16 + half * 8 + j) * F + col] = acc[nt][j];
        }
    }
}

// ---------------- self-loop term + bias (non-atomic init of agg) ----------------
// agg[i] = t[i] * dinv[i]^2 + b   ; 32 threads (one wave) per node, float4 each
__global__ void k_self_bias(const float* __restrict__ t, const float* __restrict__ dinv,
                            const float* __restrict__ bias, float* __restrict__ agg) {
    int idx  = blockIdx.x * blockDim.x + threadIdx.x;
    int node = idx >> 5;
    int q    = (idx & 31) * 4;
    if (node >= NN) return;
    float d = dinv[node];
    float s = d * d;
    float4 v  = *(const float4*)(t + (size_t)node * F + q);
    float4 bv = *(const float4*)(bias + q);
    float4 r  = make_float4(v.x * s + bv.x, v.y * s + bv.y, v.z * s + bv.z, v.w * s + bv.w);
    *(float4*)(agg + (size_t)node * F + q) = r;
}

// ---------------- edge scatter: one wave per edge, float4 per lane ----------------
__global__ __launch_bounds__(256) void k_edge_scatter(const long long* __restrict__ srcI,
                                                      const long long* __restrict__ dstI,
                                                      const float* __restrict__ t,
                                                      const float* __restrict__ dinv,
                                                      float* __restrict__ agg) {
    int gid  = blockIdx.x * blockDim.x + threadIdx.x;
    int e    = gid >> 5;
    int lane = threadIdx.x & 31;
    if (e >= NE) return;
    int s = (int)srcI[e];
    int d = (int)dstI[e];
    float nrm = dinv[s] * dinv[d];
    int q = lane * 4;
    float4 v = *(const float4*)(t + (size_t)s * F + q);
    float* ap = agg + (size_t)d * F + q;
    atomic_add_f32(ap + 0, v.x * nrm);
    atomic_add_f32(ap + 1, v.y * nrm);
    atomic_add_f32(ap + 2, v.z * nrm);
    atomic_add_f32(ap + 3, v.w * nrm);
}

__global__ void k_zero(float* p, int n) {
    int i = blockIdx.x * blockDim.x + threadIdx.x;
    if (i < n) p[i] = 0.0f;
}

// ---------------- mean pool (scatter into per-graph sums + counts) ----------------
__global__ void k_pool(const long long* __restrict__ batch, const float* __restrict__ h,
                       float* __restrict__ psum, float* __restrict__ pcnt) {
    int idx  = blockIdx.x * blockDim.x + threadIdx.x;
    int node = idx >> 5;
    int lane = idx & 31;
    if (node >= NN) return;
    int g = (int)batch[node];
    int q = lane * 4;
    float4 v = *(const float4*)(h + (size_t)node * F + q);
    float* pp = psum + (size_t)g * F + q;
    atomic_add_f32(pp + 0, v.x);
    atomic_add_f32(pp + 1, v.y);
    atomic_add_f32(pp + 2, v.z);
    atomic_add_f32(pp + 3, v.w);
    if (lane == 0) atomic_add_f32(&pcnt[g], 1.0f);
}

// ---------------- classifier: out[g,c] = (psum[g]/max(cnt,1)) @ Wc + bc ----------------
__global__ void k_final(const float* __restrict__ psum, const float* __restrict__ pcnt,
                        const float* __restrict__ Wc, const float* __restrict__ bc,
                        float* __restrict__ out) {
    int idx = blockIdx.x * blockDim.x + threadIdx.x;
    if (idx >= NG * NC) return;
    int g = idx / NC, c = idx % NC;
    float inv = 1.0f / fmaxf(pcnt[g], 1.0f);
    float acc = 0.0f;
    for (int f = 0; f < F; ++f) acc += psum[(size_t)g * F + f] * Wc[f * NC + c];
    out[idx] = acc * inv + bc[c];
}

extern "C" void kernel_launch(void* const* d_in, const int* in_sizes, int n_in,
                              void* d_out, int out_size, void* d_ws, size_t ws_size,
                              hipStream_t stream) {
    const float*     x    = (const float*)d_in[0];
    const long long* ei   = (const long long*)d_in[1];   // [2, NE] flat
    const long long* bat  = (const long long*)d_in[2];
    const float*     W1   = (const float*)d_in[3];
    const float*     b1   = (const float*)d_in[4];
    const float*     W2   = (const float*)d_in[5];
    const float*     b2   = (const float*)d_in[6];
    const float*     W3   = (const float*)d_in[7];
    const float*     b3   = (const float*)d_in[8];
    const float*     Wc   = (const float*)d_in[9];
    const float*     bc   = (const float*)d_in[10];
    float* out = (float*)d_out;

    const long long* srcI = ei;            // edge_index[0]
    const long long* dstI = ei + NE;       // edge_index[1]

    // workspace layout
    float* bufA = (float*)d_ws;                       // [NN, F] GEMM output t
    float* bufB = bufA + (size_t)NN * F;              // [NN, F] agg / h
    float* dinv = bufB + (size_t)NN * F;              // [NN]  deg -> dinv in place
    float* psum = dinv + NN;                          // [NG, F]
    float* pcnt = psum + (size_t)NG * F;              // [NG]

    const int T = 256;
    // degrees -> dinv
    k_init_deg<<<(NN + T - 1) / T, T, 0, stream>>>(dinv);
    k_deg_edges<<<(NE + T - 1) / T, T, 0, stream>>>(dstI, dinv);
    k_dinv<<<(NN + T - 1) / T, T, 0, stream>>>(dinv);

    const int gemm_blocks = (MT + 7) / 8;
    const int nodewave_blocks = (NN * 32 + T - 1) / T;
    const int edge_blocks = (int)(((long long)NE * 32 + T - 1) / T);

    // layer 1: x -> bufA -> bufB        (ReLU deferred into next GEMM's A-load)
    k_gemm_wmma<false><<<gemm_blocks, T, 0, stream>>>(x, W1, bufA);
    k_self_bias<<<nodewave_blocks, T, 0, stream>>>(bufA, dinv, b1, bufB);
    k_edge_scatter<<<edge_blocks, T, 0, stream>>>(srcI, dstI, bufA, dinv, bufB);

    // layer 2: relu(bufB) -> bufA -> bufB
    k_gemm_wmma<true><<<gemm_blocks, T, 0, stream>>>(bufB, W2, bufA);
    k_self_bias<<<nodewave_blocks, T, 0, stream>>>(bufA, dinv, b2, bufB);
    k_edge_scatter<<<edge_blocks, T, 0, stream>>>(srcI, dstI, bufA, dinv, bufB);

    // layer 3: relu(bufB) -> bufA -> bufB  (no ReLU on layer-3 output)
    k_gemm_wmma<true><<<gemm_blocks, T, 0, stream>>>(bufB, W3, bufA);
    k_self_bias<<<nodewave_blocks, T, 0, stream>>>(bufA, dinv, b3, bufB);
    k_edge_scatter<<<edge_blocks, T, 0, stream>>>(srcI, dstI, bufA, dinv, bufB);

    // mean pool + classifier
    k_zero<<<(NG * F + NG + T - 1) / T, T, 0, stream>>>(psum, NG * F + NG); // psum & pcnt contiguous
    k_pool<<<nodewave_blocks, T, 0, stream>>>(bat, bufB, psum, pcnt);
    k_final<<<(NG * NC + T - 1) / T, T, 0, stream>>>(psum, pcnt, Wc, bc, out);
}